// MoERouter_28484223107687
// MI455X (gfx1250) — compile-verified
//
#include <hip/hip_runtime.h>
#include <hip/hip_bf16.h>
#include <math.h>

// ---------------------------------------------------------------------------
// MoE router for MI455X (gfx1250, wave32).
//  - Phase 1: router logits via V_WMMA_F32_16X16X4_F32 (native fp32 WMMA).
//    W^T is staged zero-padded (16 expert rows) in LDS so the inner K-loop is
//    branch-free: 2x global_load_b64 (A) + 1x ds_load_b64 (B) + 2x WMMA.
//  - Phase 2: per-token softmax / top-2 / stats, per-block partial sums.
//  - Phase 3: deterministic final reduction of block partials -> scalars.
// ---------------------------------------------------------------------------

typedef __attribute__((ext_vector_type(2))) float v2f;
typedef __attribute__((ext_vector_type(4))) float v4f;
typedef __attribute__((ext_vector_type(8))) float v8f;

#define MOE_D 1024
#define MOE_E 8
#define MOE_K 2
#define WAVES_PER_BLOCK 4
#define TILES_PER_WAVE 2
#define P2_THREADS 256
#define WS_STRIDE 20   // floats per block partial: 8 counts + 8 prob sums + 1 ent

// ---------------------------------------------------------------------------
// Phase 1: logits[t][e] = sum_k x[t][k] * W[e][k], via fp32 WMMA.
//
// A (16x4, MxK) lane layout: lanes 0-15 -> rows M=0..15, K = {0,1};
//                            lanes 16-31 -> rows M=0..15, K = {2,3}.
// B (4x16, KxN) lane layout: VGPR j, lanes 0-15 -> K = j,   N = lane;
//                            VGPR j, lanes 16-31 -> K = j+2, N = lane-16.
// Both reduce to a contiguous float2 fetch at offset (k0 + 2*half) within the
// row; B comes from the zero-padded LDS copy of W (rows 8..15 are zeros).
// D (16x16) layout: VGPR v holds M = v (lanes 0-15) / M = v+8 (lanes 16-31),
//                   N = lane & 15.
// ---------------------------------------------------------------------------
__global__ __launch_bounds__(32 * WAVES_PER_BLOCK)
void moe_logits_wmma(const float* __restrict__ x,
                     const float* __restrict__ W,
                     float* __restrict__ logits,
                     int num_tokens) {
  __shared__ float Wpad[16 * MOE_D];   // 64 KB: experts 0..7 real, 8..15 zero

  // Cooperative stage of W^pad (once per block; amortized over the K loop).
  for (int i = threadIdx.x; i < MOE_E * MOE_D; i += blockDim.x) {
    Wpad[i] = W[i];
    Wpad[MOE_E * MOE_D + i] = 0.0f;
  }
  __syncthreads();

  const int lane = threadIdx.x & 31;
  const int wave = threadIdx.x >> 5;
  const int half = lane >> 4;      // 0: K pair {0,1} ; 1: K pair {2,3}
  const int m    = lane & 15;      // A row (token) / B column (expert)

  const int tile0 = (blockIdx.x * WAVES_PER_BLOCK + wave) * TILES_PER_WAVE;
  const int row0  = tile0 * 16;
  if (row0 >= num_tokens) return;

  const float* __restrict__ xrow0 = x + (size_t)(row0 + m) * MOE_D + 2 * half;
  const float* __restrict__ xrow1 = xrow0 + (size_t)16 * MOE_D;
  const float* __restrict__ brow  = Wpad + m * MOE_D + 2 * half;

  v8f acc0 = {};
  v8f acc1 = {};

#pragma unroll 8
  for (int k0 = 0; k0 < MOE_D; k0 += 4) {
    v2f a0 = *(const v2f*)(xrow0 + k0);   // global_load_b64, offset folded
    v2f a1 = *(const v2f*)(xrow1 + k0);   // global_load_b64
    v2f b  = *(const v2f*)(brow + k0);    // ds_load_b64 (uniform, branch-free)
    acc0 = __builtin_amdgcn_wmma_f32_16x16x4_f32(
        false, a0, false, b, (short)0, acc0, false, false);
    acc1 = __builtin_amdgcn_wmma_f32_16x16x4_f32(
        false, a1, false, b, (short)0, acc1, false, false);
  }

  // Store: lane (half,m) holds logits for tokens M = v + 8*half, expert N = m.
  if (m < MOE_E) {
    float* __restrict__ l0 =
        logits + (size_t)(row0 + 8 * half) * MOE_E + m;
    float* __restrict__ l1 = l0 + (size_t)16 * MOE_E;
#pragma unroll
    for (int v = 0; v < 8; ++v) {
      l0[(size_t)v * MOE_E] = acc0[v];
      l1[(size_t)v * MOE_E] = acc1[v];
    }
  }
}

// ---------------------------------------------------------------------------
// Phase 2: per-token softmax, entropy, top-2; per-block partial statistics.
// ---------------------------------------------------------------------------
__global__ __launch_bounds__(P2_THREADS)
void moe_phase2(const float* __restrict__ logits,
                float* __restrict__ dispatch_w,   // [N,2]
                float* __restrict__ expert_idx,   // [N,2] (as float)
                float* __restrict__ ws,           // block partials
                int num_tokens) {
  __shared__ float s_cnt[MOE_E];
  __shared__ float s_ps[MOE_E];
  __shared__ float s_ent;

  const int tid = threadIdx.x;
  if (tid < MOE_E) { s_cnt[tid] = 0.0f; s_ps[tid] = 0.0f; }
  if (tid == 0) s_ent = 0.0f;
  __syncthreads();

  const int t = blockIdx.x * P2_THREADS + tid;
  if (t < num_tokens) {
    v4f l0 = *(const v4f*)(logits + (size_t)t * MOE_E);
    v4f l1 = *(const v4f*)(logits + (size_t)t * MOE_E + 4);
    float z[MOE_E] = {l0.x, l0.y, l0.z, l0.w, l1.x, l1.y, l1.z, l1.w};

    float mx = z[0];
#pragma unroll
    for (int e = 1; e < MOE_E; ++e) mx = fmaxf(mx, z[e]);

    float p[MOE_E];
    float S = 0.0f, U = 0.0f;
#pragma unroll
    for (int e = 0; e < MOE_E; ++e) {
      float d  = z[e] - mx;
      float ex = expf(d);
      p[e] = ex;
      S += ex;
      U += ex * d;
    }
    const float invS = 1.0f / S;
    // H = log(S) - sum(p * (z - mx)) / S   (stable entropy of softmax)
    const float ent = logf(S) - U * invS;

    // top-2 with jax.lax.top_k tie-breaking (earlier index wins)
    float m1 = -1.0f; int i1 = 0;
    float m2 = -1.0f; int i2 = 0;
#pragma unroll
    for (int e = 0; e < MOE_E; ++e) {
      float pe = p[e];
      if (pe > m1)      { m2 = m1; i2 = i1; m1 = pe; i1 = e; }
      else if (pe > m2) { m2 = pe; i2 = e; }
    }

    dispatch_w[(size_t)t * MOE_K + 0] = m1 * invS;
    dispatch_w[(size_t)t * MOE_K + 1] = m2 * invS;
    expert_idx[(size_t)t * MOE_K + 0] = (float)i1;
    expert_idx[(size_t)t * MOE_K + 1] = (float)i2;

#pragma unroll
    for (int e = 0; e < MOE_E; ++e) atomicAdd(&s_ps[e], p[e] * invS);
    atomicAdd(&s_cnt[i1], 1.0f);
    atomicAdd(&s_cnt[i2], 1.0f);
    atomicAdd(&s_ent, ent);
  }
  __syncthreads();

  float* wsb = ws + (size_t)blockIdx.x * WS_STRIDE;
  if (tid < MOE_E) {
    wsb[tid]         = s_cnt[tid];
    wsb[MOE_E + tid] = s_ps[tid];
  }
  if (tid == 0) wsb[2 * MOE_E] = s_ent;
}

// ---------------------------------------------------------------------------
// Phase 3: deterministic reduction of block partials -> scalar outputs.
// out_scalars: [0]=load_balance_loss [1]=entropy_loss [2]=num_dropped
//              [3..10]=expert_usage
// ---------------------------------------------------------------------------
__global__ void moe_finalize(const float* __restrict__ ws,
                             int nblocks,
                             float* __restrict__ out_scalars,
                             int num_tokens) {
  if (blockIdx.x != 0 || threadIdx.x != 0) return;

  float cnt[MOE_E], ps[MOE_E], ent = 0.0f;
#pragma unroll
  for (int e = 0; e < MOE_E; ++e) { cnt[e] = 0.0f; ps[e] = 0.0f; }

  for (int b = 0; b < nblocks; ++b) {
    const float* wsb = ws + (size_t)b * WS_STRIDE;
#pragma unroll
    for (int e = 0; e < MOE_E; ++e) {
      cnt[e] += wsb[e];
      ps[e]  += wsb[MOE_E + e];
    }
    ent += wsb[2 * MOE_E];
  }

  // per_expert_capacity = max(1, (N * k * 1.25) // E)  (1.25 == 5/4 exact)
  long long total_cap = (long long)num_tokens * MOE_K * 5LL / 4LL;
  long long cap = total_cap / MOE_E;
  if (cap < 1) cap = 1;
  const float capf = (float)cap;
  const float invN = 1.0f / (float)num_tokens;

  float lbl = 0.0f, dropped = 0.0f;
#pragma unroll
  for (int e = 0; e < MOE_E; ++e) {
    float avg_p = ps[e] * invN;             // avg_probs[e]
    lbl += (cnt[e] * invN) * avg_p;
    if (cnt[e] > capf) dropped += 1.0f;
  }
  lbl *= (float)MOE_E;

  float entropy = ent * invN;
  float ent_loss = logf((float)MOE_E) - entropy;
  if (ent_loss < 0.0f) ent_loss = 0.0f;

  out_scalars[0] = lbl;
  out_scalars[1] = ent_loss;
  out_scalars[2] = dropped;
#pragma unroll
  for (int e = 0; e < MOE_E; ++e) out_scalars[3 + e] = cnt[e] * invN;
}

// ---------------------------------------------------------------------------
extern "C" void kernel_launch(void* const* d_in, const int* in_sizes, int n_in,
                              void* d_out, int out_size, void* d_ws, size_t ws_size,
                              hipStream_t stream) {
  const float* x = (const float*)d_in[0];   // [8,4096,1024] fp32
  const float* W = (const float*)d_in[1];   // [8,1024] fp32
  // d_in[2] = force_top_k (==2 in this configuration; layout depends on it)

  const int N = in_sizes[0] / MOE_D;        // 32768 tokens

  float* out        = (float*)d_out;
  float* dispatch_w = out;                          // N*2
  float* expert_idx = out + (size_t)N * 2;          // N*2
  float* logits     = out + (size_t)N * 4;          // N*8
  float* scalars    = out + (size_t)N * 12;         // 3 + 8
  float* ws         = (float*)d_ws;

  // Phase 1: WMMA logits. Two 16-token tiles per wave, W^pad in LDS.
  const int tiles   = N / 16;                                      // 2048
  const int wtiles  = tiles / TILES_PER_WAVE;                      // 1024
  const int blocksA = wtiles / WAVES_PER_BLOCK;                    // 256
  moe_logits_wmma<<<blocksA, 32 * WAVES_PER_BLOCK, 0, stream>>>(x, W, logits, N);

  // Phase 2: softmax / top-2 / per-block stats.
  const int blocksB = (N + P2_THREADS - 1) / P2_THREADS;           // 128
  moe_phase2<<<blocksB, P2_THREADS, 0, stream>>>(logits, dispatch_w, expert_idx,
                                                 ws, N);

  // Phase 3: final scalars.
  moe_finalize<<<1, 32, 0, stream>>>(ws, blocksB, scalars, N);
}